// ColBERT_84619445666711
// MI455X (gfx1250) — compile-verified
//
#include <hip/hip_runtime.h>
#include <hip/hip_bf16.h>
#include <math.h>
#include <stdint.h>

typedef __attribute__((ext_vector_type(2))) float v2f;
typedef __attribute__((ext_vector_type(8))) float v8f;
typedef __attribute__((ext_vector_type(4))) unsigned int u32x4;
typedef __attribute__((ext_vector_type(8))) int          i32x8;
typedef __attribute__((ext_vector_type(4))) int          i32x4;

#define NQ    35
#define LP    180
#define HDIM  768
#define DDIM  128
#define NEGV  (-1e30f)

// ---- CDNA5 dependency-counter waits ---------------------------------------
__device__ __forceinline__ void wait_async0() {
#if __has_builtin(__builtin_amdgcn_s_wait_asynccnt)
    __builtin_amdgcn_s_wait_asynccnt(0);
#else
    asm volatile("s_wait_asynccnt 0x0" ::: "memory");
#endif
}
__device__ __forceinline__ void wait_tensor0() {
#if __has_builtin(__builtin_amdgcn_s_wait_tensorcnt)
    __builtin_amdgcn_s_wait_tensorcnt(0);
#else
    asm volatile("s_wait_tensorcnt 0x0" ::: "memory");
#endif
}

// ---- async copy of one 768x16 W panel into LDS (ASYNCcnt path) ------------
// 3072 16-byte chunks; 12 per thread at 256 threads. LDS layout [k][16].
__device__ __forceinline__ void issue_wpanel_async(const float* __restrict__ W,
                                                   int ctn, float* bufptr, int tid)
{
    #pragma unroll
    for (int j = 0; j < 12; ++j) {
        const int chunk = tid + j * 256;
        const int k  = chunk >> 2;
        const int d4 = (chunk & 3) << 2;
        unsigned lds = (unsigned)(uintptr_t)(bufptr + k * 16 + d4);
        unsigned long long g =
            (unsigned long long)(uintptr_t)(W + (size_t)k * DDIM + ctn * 16 + d4);
        asm volatile("global_load_async_to_lds_b128 %0, %1, off"
                     :: "v"(lds), "v"(g) : "memory");
    }
}

// ---------------------------------------------------------------------------
// Kernel 1: X = H[b] @ W + bias, normalize each column over the sequence axis.
// fp32 WMMA 16x16x4; W panels double-buffered via async-to-LDS so the 49 KB
// panel load overlaps the 192-WMMA compute phase. One barrier per tile.
// ---------------------------------------------------------------------------
__global__ __launch_bounds__(256)
void proj_norm_kernel(const float* __restrict__ Hh, const float* __restrict__ W,
                      const float* __restrict__ bias, float* __restrict__ Xout,
                      int L)
{
    __shared__ float wpanel[2][HDIM * 16];  // 2 x 49 KB double buffer
    __shared__ float colpart[2][12 * 16];   // per-rowtile sum-of-squares partials

    const int tid  = threadIdx.x;
    const int wave = tid >> 5;
    const int lane = tid & 31;
    const int dc    = lane & 15;
    const int khalf = (lane >> 4) * 2;
    const int hi8   = (lane >> 4) ? 8 : 0;
    const int nRT   = (L + 15) >> 4;

    const float* Hb = Hh + (size_t)blockIdx.x * L * HDIM;
    float*       Xb = Xout + (size_t)blockIdx.x * L * DDIM;

    const int rt0 = wave, rt1 = wave + 8;
    const bool has0 = (rt0 < nRT), has1 = (rt1 < nRT);
    int row0 = rt0 * 16 + dc; if (row0 > L - 1) row0 = L - 1;   // clamp: padded
    int row1 = rt1 * 16 + dc; if (row1 > L - 1) row1 = L - 1;   // rows discarded
    const float* arow0 = Hb + (size_t)row0 * HDIM + khalf;
    const float* arow1 = Hb + (size_t)row1 * HDIM + khalf;

    issue_wpanel_async(W, 0, wpanel[0], tid);
    wait_async0();
    __syncthreads();

    for (int ct = 0; ct < 8; ++ct) {
        const int bufi = ct & 1;
        if (ct < 7) issue_wpanel_async(W, ct + 1, wpanel[1 - bufi], tid);

        const float* wp = wpanel[bufi];
        v8f c0 = {}, c1 = {};
        if (has0) {
            if (has1) {
                #pragma unroll 4
                for (int k0 = 0; k0 < HDIM; k0 += 4) {
                    const int kk = k0 + khalf;
                    v2f bv;
                    bv.x = wp[kk * 16 + dc];
                    bv.y = wp[(kk + 1) * 16 + dc];
                    v2f a0 = *(const v2f*)(arow0 + k0);
                    v2f a1 = *(const v2f*)(arow1 + k0);
                    c0 = __builtin_amdgcn_wmma_f32_16x16x4_f32(
                            false, a0, false, bv, (short)0, c0, false, false);
                    c1 = __builtin_amdgcn_wmma_f32_16x16x4_f32(
                            false, a1, false, bv, (short)0, c1, false, false);
                }
            } else {
                #pragma unroll 4
                for (int k0 = 0; k0 < HDIM; k0 += 4) {
                    const int kk = k0 + khalf;
                    v2f bv;
                    bv.x = wp[kk * 16 + dc];
                    bv.y = wp[(kk + 1) * 16 + dc];
                    v2f a0 = *(const v2f*)(arow0 + k0);
                    c0 = __builtin_amdgcn_wmma_f32_16x16x4_f32(
                            false, a0, false, bv, (short)0, c0, false, false);
                }
            }
            const float bb = bias[ct * 16 + dc];
            float ps0 = 0.0f, ps1 = 0.0f;
            #pragma unroll
            for (int r = 0; r < 8; ++r) {
                float v0 = c0[r] + bb;  c0[r] = v0;
                if (rt0 * 16 + r + hi8 < L) ps0 += v0 * v0;
                float v1 = c1[r] + bb;  c1[r] = v1;
                if (rt1 * 16 + r + hi8 < L) ps1 += v1 * v1;
            }
            ps0 += __shfl_xor(ps0, 16, 32);
            ps1 += __shfl_xor(ps1, 16, 32);
            if (lane < 16) {
                colpart[bufi][rt0 * 16 + dc] = ps0;
                if (has1) colpart[bufi][rt1 * 16 + dc] = ps1;
            }
        }
        wait_async0();          // my async panel writes are in LDS
        __syncthreads();        // colpart + next panel visible block-wide

        float s = 0.0f;
        for (int rt = 0; rt < nRT; ++rt) s += colpart[bufi][rt * 16 + dc];
        const float rn = 1.0f / fmaxf(sqrtf(s), 1e-12f);

        if (has0) {
            #pragma unroll
            for (int r = 0; r < 8; ++r) {
                int rr0 = rt0 * 16 + r + hi8;
                if (rr0 < L) Xb[(size_t)rr0 * DDIM + ct * 16 + dc] = c0[r] * rn;
                if (has1) {
                    int rr1 = rt1 * 16 + r + hi8;
                    if (rr1 < L) Xb[(size_t)rr1 * DDIM + ct * 16 + dc] = c1[r] * rn;
                }
            }
        }
    }
}

// ---------------------------------------------------------------------------
// Kernel 2: MaxSim for one (qb, pb) pair; fp32 WMMA 16x16x4, K=128.
// p tile staged by the Tensor Data Mover (one D#, pad_enable reproduces the
// 132-float bank-conflict-free LDS stride; rows >= 180 read back as zeros).
// q A-fragments live in 64 VGPRs and are reused across all 12 column tiles.
// ---------------------------------------------------------------------------
#define PROWS 192
#define LSTR  132         // 128 dwords data + 4 dwords pad per row

__global__ __launch_bounds__(128)
void maxsim_kernel(const float* __restrict__ qn, const float* __restrict__ pn,
                   const unsigned char* __restrict__ mask,
                   float* __restrict__ out, int colOfs)
{
    __shared__ float pt[PROWS * LSTR];    // ~101 KB
    __shared__ float rowmax[48];
    __shared__ int   smask[PROWS];

    const int qb = blockIdx.x, pb = blockIdx.y;
    const int tid = threadIdx.x, wave = tid >> 5, lane = tid & 31;
    const int nc    = lane & 15;
    const int khalf = (lane >> 4) * 2;

    const float* pbase = pn + (size_t)pb * LP * DDIM;

#if __has_builtin(__builtin_amdgcn_tensor_load_to_lds)
    if (wave == 3) {                      // idle compute wave drives the TDM
        const unsigned lds0 = (unsigned)(uintptr_t)&pt[0];
        const unsigned long long ga = (unsigned long long)(uintptr_t)pbase;
        u32x4 g0;
        g0[0] = 1u;                                   // count=1, user mode
        g0[1] = lds0;                                 // lds_addr
        g0[2] = (unsigned)(ga & 0xFFFFFFFFu);         // global_addr[31:0]
        g0[3] = (unsigned)((ga >> 32) & 0x1FFFFFFu)   // global_addr[56:32]
              | (2u << 30);                           // type = image
        i32x8 g1;
        g1[0] = (2 << 16)      // data_size = 4B
              | (1 << 20)      // pad_enable
              | (6 << 22)      // pad_interval: 128 dwords
              | (3 << 25);     // pad_amount:   4 dwords  -> stride 132
        g1[1] = (DDIM & 0xFFFF) << 16;                // tensor_dim0 lo16
        g1[2] = (LP & 0xFFFF) << 16;                  // dim0 hi=0 | tensor_dim1 lo16
        g1[3] = (DDIM & 0xFFFF) << 16;                // dim1 hi=0 | tile_dim0=128
        g1[4] = PROWS;                                // tile_dim1=192 (OOB rows -> 0)
        g1[5] = DDIM;                                 // tensor_dim0_stride = 128
        g1[6] = 0;
        g1[7] = 0;
        i32x4 gz4 = {0, 0, 0, 0};
        i32x8 gz8 = {0, 0, 0, 0, 0, 0, 0, 0};
        // 6-arg form on this toolchain: (g0, g1, g2, g3, g4, cpol)
        __builtin_amdgcn_tensor_load_to_lds(g0, g1, gz4, gz4, gz8, 0);
        wait_tensor0();
    }
#else
    for (int e = tid; e < PROWS * DDIM; e += 128) {
        int r = e >> 7, k = e & 127;
        pt[r * LSTR + k] = (r < LP) ? pbase[r * DDIM + k] : 0.0f;
    }
#endif
    for (int e = tid; e < PROWS; e += 128)
        smask[e] = (e < LP) ? (int)mask[(size_t)pb * LP + e] : 0;

    v2f aq[32];
    if (wave < 3) {                       // waves 0..2 own row tiles 0..2
        int row = wave * 16 + nc;
        if (row > NQ - 1) row = NQ - 1;   // clamped; garbage rows never summed
        const float* qrow = qn + (size_t)qb * NQ * DDIM + (size_t)row * DDIM + khalf;
        #pragma unroll
        for (int i = 0; i < 32; ++i) aq[i] = *(const v2f*)(qrow + i * 4);
    }
    __syncthreads();

    if (wave < 3) {
        const int rt = wave;
        v8f run;
        #pragma unroll
        for (int r = 0; r < 8; ++r) run[r] = NEGV;

        for (int ct = 0; ct < 12; ++ct) {
            const int token = ct * 16 + nc;
            const float* prow = pt + token * LSTR + khalf;
            v8f c = {};
            #pragma unroll
            for (int i = 0; i < 32; ++i) {
                v2f bv = *(const v2f*)(prow + i * 4);
                c = __builtin_amdgcn_wmma_f32_16x16x4_f32(
                        false, aq[i], false, bv, (short)0, c, false, false);
            }
            const bool ok = (token < LP) && (smask[token] != 0);
            #pragma unroll
            for (int r = 0; r < 8; ++r) {
                float v = ok ? c[r] : NEGV;     // select, never max vs garbage
                run[r] = fmaxf(run[r], v);
            }
        }
        #pragma unroll
        for (int r = 0; r < 8; ++r) {
            float v = run[r];
            v = fmaxf(v, __shfl_xor(v, 8, 16));
            v = fmaxf(v, __shfl_xor(v, 4, 16));
            v = fmaxf(v, __shfl_xor(v, 2, 16));
            v = fmaxf(v, __shfl_xor(v, 1, 16));
            run[r] = v;
        }
        if (lane == 0 || lane == 16) {
            const int rbase = rt * 16 + ((lane == 16) ? 8 : 0);
            #pragma unroll
            for (int r = 0; r < 8; ++r) rowmax[rbase + r] = run[r];
        }
    }
    __syncthreads();

    if (wave == 0) {
        float s = (lane < NQ) ? rowmax[lane] : 0.0f;
        if (lane < NQ - 32) s += rowmax[32 + lane];
        s += __shfl_xor(s, 16, 32);
        s += __shfl_xor(s, 8, 32);
        s += __shfl_xor(s, 4, 32);
        s += __shfl_xor(s, 2, 32);
        s += __shfl_xor(s, 1, 32);
        if (lane == 0) out[(size_t)qb * 192 + colOfs + pb] = s;
    }
}

// ---------------------------------------------------------------------------
extern "C" void kernel_launch(void* const* d_in, const int* in_sizes, int n_in,
                              void* d_out, int out_size, void* d_ws, size_t ws_size,
                              hipStream_t stream) {
    const float*         q_h   = (const float*)d_in[0];
    const float*         p_h   = (const float*)d_in[1];
    const float*         n_h   = (const float*)d_in[2];
    const float*         W     = (const float*)d_in[3];
    const float*         bias  = (const float*)d_in[4];
    const unsigned char* pmask = (const unsigned char*)d_in[5];
    const unsigned char* nmask = (const unsigned char*)d_in[6];
    float* out = (float*)d_out;

    float* ws   = (float*)d_ws;
    float* qn   = ws;                              //  96*35*128
    float* posn = qn   + 96 * NQ * DDIM;           //  96*180*128
    float* negn = posn + 96 * LP * DDIM;           //  96*180*128

    proj_norm_kernel<<<96, 256, 0, stream>>>(q_h, W, bias, qn,   NQ);
    proj_norm_kernel<<<96, 256, 0, stream>>>(p_h, W, bias, posn, LP);
    proj_norm_kernel<<<96, 256, 0, stream>>>(n_h, W, bias, negn, LP);

    dim3 grid(96, 96);
    maxsim_kernel<<<grid, 128, 0, stream>>>(qn, posn, pmask, out, 0);
    maxsim_kernel<<<grid, 128, 0, stream>>>(qn, negn, nmask, out, 96);
}